// MoELayer_56435870269504
// MI455X (gfx1250) — compile-verified
//
#include <hip/hip_runtime.h>
#include <math.h>

// ---------------------------------------------------------------------------
// MoE (top-2, SwiGLU) for MI455X / gfx1250.
// Memory-bound (567 GFLOP useful vs ~345MB min HBM traffic => ~15us floor at
// 23.3 TB/s, while even f32 WMMA gives a ~2us compute ceiling). So: exact f32
// V_WMMA_F32_16X16X4_F32 (no bf16 conversion error), gathered top-2 dispatch
// (4x FLOP saving vs dense), per-expert weights L2-resident (34.6MB < 192MB),
// async global->LDS double-buffered staging tracked by ASYNCcnt.
// ---------------------------------------------------------------------------

typedef float v2f __attribute__((ext_vector_type(2)));
typedef float v8f __attribute__((ext_vector_type(8)));
typedef int   v4i __attribute__((ext_vector_type(4)));

#define S_TOK  16384      // 4 * 4096 tokens
#define DM     1024       // d_model
#define HID    2816       // hidden
#define NEXP   8
#define NC     256        // hidden chunk (2816 = 11 * 256): 16 n-tiles = 8 waves x 2
#define NCHUNK 11
#define TILE_M 32

// Async global->LDS staging (GLOBAL_LOAD_ASYNC_TO_LDS_B128 + ASYNCcnt), with
// compile-safe fallback to synchronous float4 staging if the builtin is absent.
#if defined(__has_builtin)
#  if __has_builtin(__builtin_amdgcn_global_load_async_to_lds_b128)
#    define USE_ASYNC_LDS 1
#  endif
#endif
#ifndef USE_ASYNC_LDS
#  define USE_ASYNC_LDS 0
#endif

// Builtin signature (from compiler diagnostic): param0 = AS1 (global) v4i*,
// param1 = AS3 (LDS) v4i*, then imm offset, imm cpol.
typedef __attribute__((address_space(1))) v4i g_v4i;
typedef __attribute__((address_space(3))) v4i l_v4i;

// D = A(16x4,f32) * B(4x16,f32) + C(16x16,f32)
// Fragment layouts per CDNA5 ISA 7.12.2:
//   A: lanes 0-15 hold (M=lane, K={0,1}); lanes 16-31 hold K={2,3}.
//   B (mirror): lanes 0-15 hold (K={0,1}, N=lane); lanes 16-31 K={2,3}.
//   C/D: lane gives N=lane%16; VGPR v gives M = v + 8*(lane/16).
__device__ __forceinline__ v8f wmma4(v2f a, v2f b, v8f c) {
  return __builtin_amdgcn_wmma_f32_16x16x4_f32(false, a, false, b,
                                               (short)0, c, false, false);
}

__device__ __forceinline__ float silu(float v) {
  return v / (1.0f + __expf(-v));
}

// ---------------------------------------------------------------------------
// Kernel 1: zero output accumulator + dispatch counters.
// ---------------------------------------------------------------------------
__global__ __launch_bounds__(256)
void zero_kernel(float* __restrict__ out, int* __restrict__ cnt) {
  size_t i = ((size_t)blockIdx.x * 256 + threadIdx.x) * 8;  // 8192 blocks cover 16.78M floats
  float4 z = make_float4(0.f, 0.f, 0.f, 0.f);
  *(float4*)(out + i)     = z;
  *(float4*)(out + i + 4) = z;
  if (blockIdx.x == 0 && threadIdx.x < NEXP) cnt[threadIdx.x] = 0;
}

// ---------------------------------------------------------------------------
// Kernel 2: router. One wave32 per token: wave xor-reduced logits, strict->
// top-2 (matches jax top_k tie-break), 2-way softmax, atomic append into
// per-expert dispatch lists.
// ---------------------------------------------------------------------------
__global__ __launch_bounds__(256)
void router_kernel(const float* __restrict__ x, const float* __restrict__ rw,
                   int* __restrict__ cnt, int* __restrict__ idx_list,
                   float* __restrict__ gate_list) {
  const int wid  = threadIdx.x >> 5;
  const int lane = threadIdx.x & 31;
  const int t    = blockIdx.x * 8 + wid;

  float acc[NEXP];
#pragma unroll
  for (int e = 0; e < NEXP; ++e) acc[e] = 0.f;

  const float* xr = x + (size_t)t * DM;
  for (int d = lane; d < DM; d += 32) {
    float xv = xr[d];
    const float* r = rw + (size_t)d * NEXP;
#pragma unroll
    for (int e = 0; e < NEXP; ++e) acc[e] += xv * r[e];
  }
#pragma unroll
  for (int off = 16; off > 0; off >>= 1) {
#pragma unroll
    for (int e = 0; e < NEXP; ++e) acc[e] += __shfl_xor(acc[e], off, 32);
  }

  if (lane == 0) {
    int e0 = 0; float v0 = acc[0];
#pragma unroll
    for (int e = 1; e < NEXP; ++e) if (acc[e] > v0) { v0 = acc[e]; e0 = e; }
    int e1 = -1; float v1 = -3.0e38f;
#pragma unroll
    for (int e = 0; e < NEXP; ++e)
      if (e != e0 && acc[e] > v1) { v1 = acc[e]; e1 = e; }
    float p0 = 1.0f / (1.0f + __expf(v1 - v0));   // softmax over top-2 logits
    float p1 = 1.0f - p0;
    int s0 = atomicAdd(&cnt[e0], 1);
    idx_list[e0 * S_TOK + s0]  = t;
    gate_list[e0 * S_TOK + s0] = p0;
    int s1 = atomicAdd(&cnt[e1], 1);
    idx_list[e1 * S_TOK + s1]  = t;
    gate_list[e1 * S_TOK + s1] = p1;
  }
}

// ---------------------------------------------------------------------------
// Kernel 3: fused expert FFN over gathered tokens.
// grid = (S/32 tiles, 11 hidden chunks, 8 experts); 256 thr = 8 wave32.
// Phase 1: h[32 x 256] = silu(X@W1) * (X@W3), K=1024, one K sweep, each wave
//          owns 2 n-tiles; X tile async-staged (double-buffered) into LDS.
// Phase 2: out += gate * (h @ W2), K=256, scatter via global_atomic_add_f32
//          (output is L2-resident: 64MB < 192MB L2).
// ---------------------------------------------------------------------------
__global__ __launch_bounds__(256)
void moe_ffn_kernel(const float* __restrict__ x,
                    const float* __restrict__ w1,
                    const float* __restrict__ w3,
                    const float* __restrict__ w2,
                    const int*   __restrict__ cnt,
                    const int*   __restrict__ idx_list,
                    const float* __restrict__ gate_list,
                    float* __restrict__ out) {
  __shared__ float As[2][TILE_M][36];    // 144B rows: 16B aligned, conflict-free A-frag reads
  __shared__ float hs[TILE_M][NC + 3];   // stride 259: both half-wave frag groups conflict-free
  __shared__ int   tcache[TILE_M];
  __shared__ float gcache[TILE_M];

  const int e       = blockIdx.z;
  const int chunk   = blockIdx.y;
  const int rowbase = blockIdx.x * TILE_M;
  const int ne      = cnt[e];
  if (rowbase >= ne) return;             // uniform early-exit (before any barrier)

  const int tid  = threadIdx.x;
  const int wid  = tid >> 5;
  const int lane = tid & 31;
  const int lm   = lane & 15;
  const int koff = (lane >> 4) * 2;      // K sub-offset within a 4-step: 0 or 2
  const int mhi  = (lane >> 4) * 8;      // C/D row offset: 0 or 8

  const float* w1e = w1 + (size_t)e * DM * HID;
  const float* w3e = w3 + (size_t)e * DM * HID;
  const float* w2e = w2 + (size_t)e * HID * DM;

  if (tid < TILE_M) {
    int grow   = rowbase + tid;
    bool valid = grow < ne;
    tcache[tid] = valid ? idx_list[e * S_TOK + grow]  : 0;    // padded rows read token 0
    gcache[tid] = valid ? gate_list[e * S_TOK + grow] : 0.f;  // ...but contribute gate 0
  }
  __syncthreads();

  // Per-thread staging slice: row r = tid/8, 4 floats at col c = (tid%8)*4.
  const int sr = tid >> 3;
  const int sc = (tid & 7) * 4;
  const float* srow = x + (size_t)tcache[sr] * DM + sc;

  // ------------- Phase 1: h = silu(X@W1) * (X@W3), single K sweep -----------
  const int colb0 = chunk * NC + wid * 16;        // n-tile wid (tile wid+8 = +128)

  v8f acc1a0 = {0,0,0,0,0,0,0,0}, acc1b0 = {0,0,0,0,0,0,0,0};
  v8f acc3a0 = {0,0,0,0,0,0,0,0}, acc3b0 = {0,0,0,0,0,0,0,0};
  v8f acc1a1 = {0,0,0,0,0,0,0,0}, acc1b1 = {0,0,0,0,0,0,0,0};
  v8f acc3a1 = {0,0,0,0,0,0,0,0}, acc3b1 = {0,0,0,0,0,0,0,0};

#if USE_ASYNC_LDS
  // Prologue: async-stage slice kk=0 into buffer 0 (ASYNCcnt-tracked).
  __builtin_amdgcn_global_load_async_to_lds_b128(
      (g_v4i*)(srow + 0), (l_v4i*)&As[0][sr][sc], 0, 0);
#endif

  for (int kk = 0; kk < DM; kk += 32) {
    const int cur = (kk >> 5) & 1;
#if USE_ASYNC_LDS
    asm volatile("s_wait_asynccnt 0" ::: "memory");  // my slice landed
    __syncthreads();                                  // everyone's slice visible;
                                                      // all readers of buf cur^1 retired
    if (kk + 32 < DM)                                 // prefetch next slice
      __builtin_amdgcn_global_load_async_to_lds_b128(
          (g_v4i*)(srow + kk + 32), (l_v4i*)&As[cur ^ 1][sr][sc], 0, 0);
#else
    __syncthreads();
    *(float4*)&As[cur][sr][sc] = *(const float4*)(srow + kk);
    __syncthreads();
#endif
#pragma unroll
    for (int k4 = 0; k4 < 32; k4 += 4) {
      v2f a0, a1, b10, b30, b11, b31;
      a0.x = As[cur][lm][k4 + koff];      a0.y = As[cur][lm][k4 + koff + 1];
      a1.x = As[cur][lm + 16][k4 + koff]; a1.y = As[cur][lm + 16][k4 + koff + 1];
      const int kb = kk + k4 + koff;
      const float* p1 = w1e + (size_t)kb * HID + colb0 + lm;
      const float* p3 = w3e + (size_t)kb * HID + colb0 + lm;
      b10.x = p1[0];    b10.y = p1[HID];
      b30.x = p3[0];    b30.y = p3[HID];
      b11.x = p1[128];  b11.y = p1[HID + 128];
      b31.x = p3[128];  b31.y = p3[HID + 128];
      acc1a0 = wmma4(a0, b10, acc1a0);
      acc1b0 = wmma4(a1, b10, acc1b0);
      acc3a0 = wmma4(a0, b30, acc3a0);
      acc3b0 = wmma4(a1, b30, acc3b0);
      acc1a1 = wmma4(a0, b11, acc1a1);
      acc1b1 = wmma4(a1, b11, acc1b1);
      acc3a1 = wmma4(a0, b31, acc3a1);
      acc3b1 = wmma4(a1, b31, acc3b1);
    }
  }

  {  // SwiGLU epilogue -> hs
    const int c0 = wid * 16 + lm;        // n-tile wid
    const int c1 = c0 + 128;             // n-tile wid+8
#pragma unroll
    for (int v = 0; v < 8; ++v) {
      hs[v + mhi][c0]      = silu(acc1a0[v]) * acc3a0[v];
      hs[v + mhi + 16][c0] = silu(acc1b0[v]) * acc3b0[v];
      hs[v + mhi][c1]      = silu(acc1a1[v]) * acc3a1[v];
      hs[v + mhi + 16][c1] = silu(acc1b1[v]) * acc3b1[v];
    }
  }
  __syncthreads();

  // -------------------- Phase 2: out += gate * (h @ W2) ---------------------
  for (int nt2 = wid; nt2 < DM / 16; nt2 += 8) {
    v8f accA = {0,0,0,0,0,0,0,0}, accB = {0,0,0,0,0,0,0,0};
    const int nb = nt2 * 16 + lm;
#pragma unroll 4
    for (int k4 = 0; k4 < NC; k4 += 4) {
      v2f a0, a1, b;
      a0.x = hs[lm][k4 + koff];      a0.y = hs[lm][k4 + koff + 1];
      a1.x = hs[lm + 16][k4 + koff]; a1.y = hs[lm + 16][k4 + koff + 1];
      const int kb = chunk * NC + k4 + koff;
      const float* p2 = w2e + (size_t)kb * DM + nb;
      b.x = p2[0];  b.y = p2[DM];
      accA = wmma4(a0, b, accA);
      accB = wmma4(a1, b, accB);
    }
#pragma unroll
    for (int v = 0; v < 8; ++v) {
      int m0 = v + mhi;
      atomicAdd(out + (size_t)tcache[m0] * DM + nb, gcache[m0] * accA[v]);
      int m1 = m0 + 16;
      atomicAdd(out + (size_t)tcache[m1] * DM + nb, gcache[m1] * accB[v]);
    }
  }
}

// ---------------------------------------------------------------------------
extern "C" void kernel_launch(void* const* d_in, const int* in_sizes, int n_in,
                              void* d_out, int out_size, void* d_ws, size_t ws_size,
                              hipStream_t stream) {
  (void)in_sizes; (void)n_in; (void)out_size; (void)ws_size;
  const float* x  = (const float*)d_in[0];   // [4,4096,1024]
  const float* rw = (const float*)d_in[1];   // [1024,8]
  const float* w1 = (const float*)d_in[2];   // [8,1024,2816]
  const float* w3 = (const float*)d_in[3];   // [8,1024,2816]
  const float* w2 = (const float*)d_in[4];   // [8,2816,1024]
  float* out = (float*)d_out;

  char*  ws        = (char*)d_ws;            // ~1.05 MB total workspace
  int*   cnt       = (int*)ws;
  int*   idx_list  = (int*)(ws + 256);
  float* gate_list = (float*)(ws + 256 + (size_t)NEXP * S_TOK * sizeof(int));

  zero_kernel  <<<8192,      256, 0, stream>>>(out, cnt);
  router_kernel<<<S_TOK / 8, 256, 0, stream>>>(x, rw, cnt, idx_list, gate_list);

  dim3 grid(S_TOK / TILE_M, NCHUNK, NEXP);   // (512, 11, 8); inactive tiles exit early
  moe_ffn_kernel<<<grid, 256, 0, stream>>>(x, w1, w3, w2,
                                           cnt, idx_list, gate_list, out);
}